// M3GNetInteraction_66357244723541
// MI455X (gfx1250) — compile-verified
//
#include <hip/hip_runtime.h>
#include <hip/hip_bf16.h>
#include <math.h>

// ---------------------------------------------------------------------------
// M3GNet interaction block for MI455X (gfx1250, wave32, WMMA).
// All GEMMs use V_WMMA_F32_16X16X4_F32 (exact fp32, matrix pipe).
// Constants from the reference:
#define N_NODES 20000
#define N_EDGES 640000
#define N_TRIP  1000000
#define CDIM    128
#define EDIM    64
#define CUTOFF  5.0f
#define LOG2F_  0.69314718055994531f
#define PI_F    3.14159265358979323f
// gamma = 1/(2*(CUTOFF/E)^2) = 81.92
#define GAMMA_  81.92f
// centers: c_j = CUTOFF * j / (E-1)
#define CSTEP_  (5.0f / 63.0f)

typedef float v2f __attribute__((ext_vector_type(2)));
typedef float v8f __attribute__((ext_vector_type(8)));

__device__ __forceinline__ v8f wmma4(v2f a, v2f b, v8f c) {
  // D = A(16x4) * B(4x16) + C   (fp32 WMMA, wave32)
  return __builtin_amdgcn_wmma_f32_16x16x4_f32(
      /*neg_a=*/false, a, /*neg_b=*/false, b,
      /*c_mod=*/(short)0, c, /*reuse_a=*/false, /*reuse_b=*/false);
}

__device__ __forceinline__ float sspf(float x) {
  // softplus(x) - log(2), numerically stable
  float sp = fmaxf(x, 0.0f) + log1pf(expf(-fabsf(x)));
  return sp - LOG2F_;
}

__device__ __forceinline__ v8f zero8() {
  v8f z = {0.f, 0.f, 0.f, 0.f, 0.f, 0.f, 0.f, 0.f};
  return z;
}

// ---------------------------------------------------------------------------
// D[M x 128] = A[M x 128] @ B[128 x 128].  One wave per 16x16 output tile.
// 256 threads = 8 waves per block.  M must be a multiple of 16.
__global__ void gemm128_kernel(const float* __restrict__ A,
                               const float* __restrict__ B,
                               float* __restrict__ D, int M) {
  const int wave = threadIdx.x >> 5;
  const int lane = threadIdx.x & 31;
  const int tile = blockIdx.x * 8 + wave;
  const int ntiles = (M / 16) * (CDIM / 16);
  if (tile >= ntiles) return;
  const int mt = tile >> 3;        // M tile
  const int nt = tile & 7;         // N tile
  const int rowA = lane & 15;      // A row within tile
  const int kg   = (lane >> 4) * 2; // K sub-offset (0 or 2)
  const int col  = lane & 15;      // D column within tile
  const int mh   = (lane >> 4) * 8; // D row-half offset

  v8f acc = zero8();
  const float* Arow = A + (size_t)(mt * 16 + rowA) * CDIM;
  for (int k = 0; k < CDIM; k += 4) {
    v2f a;
    a.x = Arow[k + kg];
    a.y = Arow[k + kg + 1];
    v2f b;
    b.x = B[(k + kg) * CDIM + nt * 16 + col];
    b.y = B[(k + kg + 1) * CDIM + nt * 16 + col];
    acc = wmma4(a, b, acc);
  }
  for (int r = 0; r < 8; ++r)
    D[(size_t)(mt * 16 + r + mh) * CDIM + nt * 16 + col] = acc[r];
}

// ---------------------------------------------------------------------------
// Triplet path: z = ssp(tbf @ W3b1)  [16 x 64 in LDS], w3 = z @ W3b2 [16 x 128],
// msgs = h[idx1] * w3, atomic scatter into T[idx1].
// One wave handles 16 triplets.  No block-level sync (per-wave LDS regions).
__global__ void triplet_kernel(const float* __restrict__ h,
                               const int* __restrict__ trip,
                               const float* __restrict__ angles,
                               const float* __restrict__ r_ij,
                               const float* __restrict__ r_ik,
                               const float* __restrict__ W3b1,  // (3 x 64)
                               const float* __restrict__ W3b2,  // (64 x 128)
                               float* __restrict__ T) {
  __shared__ float sz[8][16 * 64];
  __shared__ int   sidx[8][16];

  const int wave = threadIdx.x >> 5;
  const int lane = threadIdx.x & 31;
  const int tile = blockIdx.x * 8 + wave;
  if (tile >= (N_TRIP / 16)) return;
  const int tbase = tile * 16;

  if (lane < 16) sidx[wave][lane] = trip[(tbase + lane) * 3 + 1];

  // Each lane computes half a row of z (32 cols).
  {
    const int row = lane >> 1;
    const int cb  = (lane & 1) * 32;
    const int t   = tbase + row;
    const float a  = r_ij[t];
    const float b  = r_ik[t];
    const float ca = cosf(angles[t]);
    for (int j = 0; j < 32; ++j) {
      const int c = cb + j;
      const float x = a * W3b1[c] + b * W3b1[64 + c] + ca * W3b1[128 + c];
      sz[wave][row * 64 + c] = sspf(x);
    }
  }
  __builtin_amdgcn_wave_barrier();

  const int rowA = lane & 15;
  const int kg   = (lane >> 4) * 2;
  const int col  = lane & 15;
  const int mh   = (lane >> 4) * 8;

  // w3 = z(16x64) @ W3b2(64x128): 8 N-tiles x 16 K-steps
  v8f acc[8];
  for (int nt = 0; nt < 8; ++nt) acc[nt] = zero8();
  for (int k = 0; k < EDIM; k += 4) {
    v2f a;
    a.x = sz[wave][rowA * 64 + k + kg];
    a.y = sz[wave][rowA * 64 + k + kg + 1];
    for (int nt = 0; nt < 8; ++nt) {
      v2f b;
      b.x = W3b2[(k + kg) * CDIM + nt * 16 + col];
      b.y = W3b2[(k + kg + 1) * CDIM + nt * 16 + col];
      acc[nt] = wmma4(a, b, acc[nt]);
    }
  }

  // msgs = h[idx1] * w3 ; scatter-add into T[idx1]  (T is L2-resident, 10 MB)
  for (int nt = 0; nt < 8; ++nt) {
    for (int r = 0; r < 8; ++r) {
      const int m    = r + mh;
      const int node = sidx[wave][m];
      const int c    = nt * 16 + col;
      const float v  = acc[nt][r] * h[(size_t)node * CDIM + c];
      atomicAdd(&T[(size_t)node * CDIM + c], v);
    }
  }
}

// ---------------------------------------------------------------------------
// Edge path: rb (16 x 64, in LDS) -> t = ssp(rb@W2b1) (LDS, reuses rb buffer)
// -> u = t @ W2b2 (16 x 128) ; val = h[nl1]*u + (e<N_NODES ? T[e] : 0);
// atomic scatter into agg[nl0].  One wave per 16 edges.
__global__ void edge_kernel(const float* __restrict__ h,
                            const float* __restrict__ dist,
                            const int* __restrict__ nl,    // [2 x N_EDGES] flat
                            const float* __restrict__ W2b1, // (64 x 64)
                            const float* __restrict__ W2b2, // (64 x 128)
                            const float* __restrict__ T,
                            float* __restrict__ agg) {
  __shared__ float sbuf[8][16 * 64];
  __shared__ int   snb[8][32];

  const int wave  = threadIdx.x >> 5;
  const int lane  = threadIdx.x & 31;
  const int tile  = blockIdx.x * 8 + wave;   // < 40000 exactly
  const int ebase = tile * 16;

  if (lane < 16) {
    snb[wave][lane]      = nl[ebase + lane];            // neighbour_list[0]
    snb[wave][16 + lane] = nl[N_EDGES + ebase + lane];  // neighbour_list[1]
  }

  // radial basis * cutoff -> LDS (each lane: half a row)
  {
    const int row = lane >> 1;
    const int cb  = (lane & 1) * 32;
    const float d = dist[ebase + row];
    float cut = 0.0f;
    if (d < CUTOFF) cut = 0.5f * (1.0f + cosf(PI_F * d / CUTOFF));
    for (int j = 0; j < 32; ++j) {
      const float c = CSTEP_ * (float)(cb + j);
      const float x = d - c;
      sbuf[wave][row * 64 + cb + j] = expf(-GAMMA_ * x * x) * cut;
    }
  }
  __builtin_amdgcn_wave_barrier();

  const int rowA = lane & 15;
  const int kg   = (lane >> 4) * 2;
  const int col  = lane & 15;
  const int mh   = (lane >> 4) * 8;

  // GEMM1: t = ssp(rb(16x64) @ W2b1(64x64)) : 4 N-tiles x 16 K-steps
  v8f acc1[4];
  for (int nt = 0; nt < 4; ++nt) acc1[nt] = zero8();
  for (int k = 0; k < EDIM; k += 4) {
    v2f a;
    a.x = sbuf[wave][rowA * 64 + k + kg];
    a.y = sbuf[wave][rowA * 64 + k + kg + 1];
    for (int nt = 0; nt < 4; ++nt) {
      v2f b;
      b.x = W2b1[(k + kg) * EDIM + nt * 16 + col];
      b.y = W2b1[(k + kg + 1) * EDIM + nt * 16 + col];
      acc1[nt] = wmma4(a, b, acc1[nt]);
    }
  }
  __builtin_amdgcn_wave_barrier();
  // all reads of rb done; overwrite LDS buffer with t = ssp(acc1)
  for (int nt = 0; nt < 4; ++nt)
    for (int r = 0; r < 8; ++r)
      sbuf[wave][(r + mh) * 64 + nt * 16 + col] = sspf(acc1[nt][r]);
  __builtin_amdgcn_wave_barrier();

  // GEMM2: u = t(16x64) @ W2b2(64x128) : 8 N-tiles x 16 K-steps
  v8f acc2[8];
  for (int nt = 0; nt < 8; ++nt) acc2[nt] = zero8();
  for (int k = 0; k < EDIM; k += 4) {
    v2f a;
    a.x = sbuf[wave][rowA * 64 + k + kg];
    a.y = sbuf[wave][rowA * 64 + k + kg + 1];
    for (int nt = 0; nt < 8; ++nt) {
      v2f b;
      b.x = W2b2[(k + kg) * CDIM + nt * 16 + col];
      b.y = W2b2[(k + kg + 1) * CDIM + nt * 16 + col];
      acc2[nt] = wmma4(a, b, acc2[nt]);
    }
  }

  // messages = h[nl1]*u + edge_msgs ; segment-sum into agg[nl0] (L2-resident)
  for (int nt = 0; nt < 8; ++nt) {
    for (int r = 0; r < 8; ++r) {
      const int m = r + mh;
      const int e = ebase + m;
      const int c = nt * 16 + col;
      float v = acc2[nt][r] * h[(size_t)snb[wave][16 + m] * CDIM + c];
      if (e < N_NODES) v += T[(size_t)e * CDIM + c];
      atomicAdd(&agg[(size_t)snb[wave][m] * CDIM + c], v);
    }
  }
}

// ---------------------------------------------------------------------------
extern "C" void kernel_launch(void* const* d_in, const int* in_sizes, int n_in,
                              void* d_out, int out_size, void* d_ws, size_t ws_size,
                              hipStream_t stream) {
  (void)in_sizes; (void)n_in; (void)out_size; (void)ws_size;
  const float* features = (const float*)d_in[0];
  const float* nbdist   = (const float*)d_in[1];
  const int*   nl       = (const int*)d_in[2];
  const int*   trip     = (const int*)d_in[3];
  const float* angles   = (const float*)d_in[4];
  const float* r_ij     = (const float*)d_in[5];
  const float* r_ik     = (const float*)d_in[6];
  const float* W_pre    = (const float*)d_in[7];
  const float* W2b1     = (const float*)d_in[8];
  const float* W2b2     = (const float*)d_in[9];
  const float* W3b1     = (const float*)d_in[10];
  const float* W3b2     = (const float*)d_in[11];
  const float* W_post   = (const float*)d_in[12];

  float* h   = (float*)d_ws;                     // 20000 x 128
  float* T   = h + (size_t)N_NODES * CDIM;       // 20000 x 128 (triplet scatter)
  float* agg = T + (size_t)N_NODES * CDIM;       // 20000 x 128

  // zero scatter targets T and agg (contiguous)
  hipMemsetAsync(T, 0, (size_t)2 * N_NODES * CDIM * sizeof(float), stream);

  // h = features @ W_pre : 10000 tiles / 8 waves = 1250 blocks
  gemm128_kernel<<<1250, 256, 0, stream>>>(features, W_pre, h, N_NODES);

  // triplet scatter into T : 62500 wave-tiles -> 7813 blocks (guarded)
  triplet_kernel<<<7813, 256, 0, stream>>>(h, trip, angles, r_ij, r_ik,
                                           W3b1, W3b2, T);

  // edge path + segment sum into agg : 40000 wave-tiles -> 5000 blocks (exact)
  edge_kernel<<<5000, 256, 0, stream>>>(h, nbdist, nl, W2b1, W2b2, T, agg);

  // out = agg @ W_post
  gemm128_kernel<<<1250, 256, 0, stream>>>(agg, W_post, (float*)d_out, N_NODES);
}